// WeightedAggregator_new_89489938580182
// MI455X (gfx1250) — compile-verified
//
#include <hip/hip_runtime.h>
#include <stdint.h>

// Weighted neighbor aggregation (gather + weighted mean).
//   features : [V=200000, D=128] fp32   (102.4 MB -> fits in 192 MB L2)
//   weights  : [B=100000, K=10]  fp32
//   neigh_idx: [B, K] int32
//   out[b,d] = sum_k w[b,k] * features[idx[b,k], d] / sum_k w[b,k]
//
// Memory-bound (~570 MB, 0.45 FLOP/byte -> ~24 us floor at 23.3 TB/s).
// Strategy: CDNA5 async gather-to-LDS (GLOBAL_LOAD_ASYNC_TO_LDS_B128, GVS
// addressing: uniform SGPR row base + constant per-lane offset), double
// buffered per wave; weights/reciprocal overlap the gather latency; output
// stored non-temporally to preserve L2 residency of the feature table.

#define K_NEIGH 10
#define DIM 128
#define WAVES_PER_BLOCK 4
#define THREADS (WAVES_PER_BLOCK * 32)
#define ROWS_PER_WAVE 8

typedef float v4f __attribute__((ext_vector_type(4)));

__global__ __launch_bounds__(THREADS)
void WeightedAggregator_gather_kernel(const float* __restrict__ features,
                                      const float* __restrict__ weights,
                                      const int*   __restrict__ nidx,
                                      float* __restrict__ out,
                                      int Brows, int wave_stride)
{
    // Per-wave double-buffered staging: [wave][buf][k][lane] = 16B each.
    // 4 * 2 * 10 * 32 * 16B = 40 KB LDS per block.
    __shared__ float4 stage[WAVES_PER_BLOCK][2][K_NEIGH][32];

    const int lane = threadIdx.x & 31;
    const int wv   = threadIdx.x >> 5;
    int wid = __builtin_amdgcn_readfirstlane(blockIdx.x * WAVES_PER_BLOCK + wv);

    const uint64_t fbase = (uint64_t)(uintptr_t)features;
    const uint32_t voff  = (uint32_t)(lane * 16);   // per-lane 16 B slice of a 512 B row

    // Issue K async gathers for one batch row into LDS buffer `buf`.
    // Indices are pre-read into registers so all scalar loads cluster under a
    // single KMcnt wait, then the gathers issue back-to-back in GVS mode
    // (SGPR row base, shared per-lane VGPR offset -> zero per-gather VALU).
    auto issue_row = [&](int row, int buf) {
        const int* irow = nidx + (size_t)row * K_NEIGH;
        int idxs[K_NEIGH];
        #pragma unroll
        for (int k = 0; k < K_NEIGH; ++k) idxs[k] = irow[k];
        #pragma unroll
        for (int k = 0; k < K_NEIGH; ++k) {
            const uint64_t srow  = fbase + ((uint64_t)(uint32_t)idxs[k] << 9);
            const uint32_t laddr = (uint32_t)(uintptr_t)&stage[wv][buf][k][lane];
            asm volatile("global_load_async_to_lds_b128 %0, %1, %2"
                         :: "v"(laddr), "v"(voff), "s"(srow) : "memory");
        }
    };

    int cur = 0;
    int b = wid;

    if (b < Brows) issue_row(b, 0);   // prologue: fill buffer 0

    for (; b < Brows; b += wave_stride) {
        const int bn = b + wave_stride;
        const bool has_next = (bn < Brows);

        if (has_next) {
            // Our ds_loads from the buffer being refilled must have retired
            // before new async writes can land in it.
            asm volatile("s_wait_dscnt 0x0" ::: "memory");
            issue_row(bn, cur ^ 1);
        }

        // Overlap with gather latency: weights + normalizer for current row.
        const float* wrow = weights + (size_t)b * K_NEIGH;
        float ws[K_NEIGH];
        float wsum = 0.f;
        #pragma unroll
        for (int k = 0; k < K_NEIGH; ++k) { ws[k] = wrow[k]; wsum += ws[k]; }
        const float inv = 1.0f / wsum;

        if (has_next) {
            // <=10 outstanding: current row's 10 gathers are done
            // (async loads complete in issue order).
            asm volatile("s_wait_asynccnt 0xa" ::: "memory");
        } else {
            asm volatile("s_wait_asynccnt 0x0" ::: "memory");
        }

        // Weighted accumulation: each lane owns d = lane*4 .. lane*4+3.
        v4f acc = {0.f, 0.f, 0.f, 0.f};
        #pragma unroll
        for (int k = 0; k < K_NEIGH; ++k) {
            const float4 f = stage[wv][cur][k][lane];   // ds_load_b128
            acc.x = fmaf(ws[k], f.x, acc.x);
            acc.y = fmaf(ws[k], f.y, acc.y);
            acc.z = fmaf(ws[k], f.z, acc.z);
            acc.w = fmaf(ws[k], f.w, acc.w);
        }
        acc *= inv;

        // Streaming write-once output: non-temporal b128 store keeps the
        // feature table resident in L2.
        __builtin_nontemporal_store(acc, (v4f*)(out + (size_t)b * DIM + lane * 4));

        cur ^= 1;
    }
}

extern "C" void kernel_launch(void* const* d_in, const int* in_sizes, int n_in,
                              void* d_out, int out_size, void* d_ws, size_t ws_size,
                              hipStream_t stream) {
    const float* features = (const float*)d_in[0];
    const float* weights  = (const float*)d_in[1];
    const int*   nidx     = (const int*)d_in[2];
    float* out = (float*)d_out;

    const int Brows = in_sizes[1] / K_NEIGH;           // weights is [B, K]
    int waves  = (Brows + ROWS_PER_WAVE - 1) / ROWS_PER_WAVE;
    int blocks = (waves + WAVES_PER_BLOCK - 1) / WAVES_PER_BLOCK;
    if (blocks < 1) blocks = 1;
    const int wave_stride = blocks * WAVES_PER_BLOCK;  // total waves in grid

    WeightedAggregator_gather_kernel<<<blocks, THREADS, 0, stream>>>(
        features, weights, nidx, out, Brows, wave_stride);
}